// VisualDecoder_6296422056356
// MI455X (gfx1250) — compile-verified
//
#include <hip/hip_runtime.h>
#include <hip/hip_bf16.h>

// ---------------------------------------------------------------------------
// VisualDecoder (deformable-DETR style) forward for MI455X / gfx1250.
//
// All dense linear algebra runs through V_WMMA_F32_16X16X4_F32 (full f32
// precision matches the f32 reference; the workload is L2-resident and
// bandwidth/latency bound, so low-precision WMMA buys nothing here).
// Wave32 (warpSize==32) is assumed throughout: __shfl_xor reductions use
// width 32; attention/LN use one wave per row.
// ---------------------------------------------------------------------------

typedef float v2f __attribute__((ext_vector_type(2)));
typedef float v8f __attribute__((ext_vector_type(8)));

// Model constants (static in the reference).
#define BSZ   2
#define TT    8
#define B2    16     // BSZ*TT
#define NQ    300
#define CC    256
#define NHD   8
#define HD    32
#define LV    4
#define PP    4
#define FFD   1024
#define NL    6
#define LEN   5440   // 64*64 + 32*32 + 16*16 + 8*8

// ---------------------------------------------------------------------------
// Generic GEMM:  D[M,N] = A[M,K] * W[K,N] + bias   (row-major, strided)
// Block: 256 threads = 8 waves, block tile 32(M) x 64(N); wave tile 16x16.
// K is always a multiple of 16 here (256 or 1024). Per 16-deep K chunk the
// block stages A (32x16) and B (16x64) in LDS with coalesced, UNPREDICATED
// global loads (addresses clamped in range; out-of-range rows are computed
// but never stored), then each wave issues 4 V_WMMA_F32_16X16X4_F32.
// LDS strides padded (18 / 65) for conflict-free reads on 64 banks.
// Optional relu; optional per-row zero mask (mask_flatten applied to ca_v).
// ---------------------------------------------------------------------------
__global__ void vd_gemm_kernel(const float* __restrict__ A, int lda,
                               const float* __restrict__ W, int ldw,
                               const float* __restrict__ bias,
                               float* __restrict__ D, int ldd,
                               int M, int K, int N, int relu,
                               const unsigned char* __restrict__ rowmask) {
  __shared__ float sA[32 * 18];   // [m][k], stride 18
  __shared__ float sB[16 * 65];   // [k][n], stride 65

  const int tid  = threadIdx.x;
  const int wid  = tid >> 5;
  const int lane = tid & 31;
  const int wm = wid & 1, wn = wid >> 1;
  const int tm0 = blockIdx.x * 32;
  const int tn0 = blockIdx.y * 64;

  // Cooperative-load coordinates (unconditional, clamped addresses).
  const int am_ld = tid >> 3;                 // 0..31   (A tile row)
  const int ak_ld = (tid & 7) << 1;           // 0,2,..14 (A tile k pair)
  const int ag_row = min(tm0 + am_ld, M - 1);
  const float* Arow = A + (long)ag_row * lda;

  // Wave-fragment coordinates.
  const int fm  = wm * 16 + (lane & 15);      // A tile row this lane reads
  const int bn  = wn * 16 + (lane & 15);      // B tile col this lane reads
  const int kof = (lane >> 4) << 1;           // 0 / 2 (A,B fragment K select)
  const int ncol = min(tn0 + bn, N - 1);

  v8f acc = {0.f, 0.f, 0.f, 0.f, 0.f, 0.f, 0.f, 0.f};

  for (int kc = 0; kc < K; kc += 16) {
    // ---- stage A: 512 floats, one b64 per thread, coalesced ----
    const v2f av = *(const v2f*)(Arow + kc + ak_ld);
    *(v2f*)&sA[am_ld * 18 + ak_ld] = av;

    // ---- stage B: 1024 floats, 4 per thread, 64-wide coalesced rows ----
#pragma unroll
    for (int i = 0; i < 4; ++i) {
      const int idx = tid + i * 256;
      const int kk = idx >> 6, n = idx & 63;
      const int gn = min(tn0 + n, N - 1);
      sB[kk * 65 + n] = W[(long)(kc + kk) * ldw + gn];
    }
    if (kc + 16 < K)   // pull next W chunk toward L2 (global_prefetch)
      __builtin_prefetch(&W[(long)(kc + 16 + (tid >> 6)) * ldw + tn0 + (tid & 63)], 0, 1);

    __syncthreads();

    // ---- 4 WMMAs per chunk, fed from LDS ----
#pragma unroll
    for (int kk = 0; kk < 16; kk += 4) {
      v2f a = *(const v2f*)&sA[fm * 18 + kk + kof];
      v2f b;
      b.x = sB[(kk + kof) * 65 + bn];
      b.y = sB[(kk + kof + 1) * 65 + bn];
      acc = __builtin_amdgcn_wmma_f32_16x16x4_f32(false, a, false, b,
                                                  (short)0, acc, false, false);
    }
    __syncthreads();
  }

  const bool nok = (tn0 + bn) < N;
  const float bv = bias ? bias[ncol] : 0.f;
  const int rbase = tm0 + wm * 16 + ((lane >> 4) << 3);  // C/D vgpr r -> row
#pragma unroll
  for (int r = 0; r < 8; ++r) {
    const int row = rbase + r;
    if (row < M && nok) {
      float v = acc[r] + bv;
      if (relu) v = fmaxf(v, 0.f);
      if (rowmask && rowmask[row]) v = 0.f;
      D[(long)row * ldd + ncol] = v;
    }
  }
}

// ---------------------------------------------------------------------------
// LayerNorm(residual): Dst = LN(A + B) with gamma/beta. One wave per 256-row.
// ---------------------------------------------------------------------------
__global__ void vd_ln_kernel(const float* __restrict__ A,
                             const float* __restrict__ B,
                             const float* __restrict__ g,
                             const float* __restrict__ be,
                             float* __restrict__ Dst, int rows) {
  const int wid = threadIdx.x >> 5, lane = threadIdx.x & 31;
  const int r = blockIdx.x * 8 + wid;
  if (r >= rows) return;
  const long base = (long)r * CC;
  float v[8];
  float s = 0.f;
#pragma unroll
  for (int j = 0; j < 8; ++j) {
    const int c = lane + j * 32;
    v[j] = A[base + c] + B[base + c];
    s += v[j];
  }
  for (int o = 16; o >= 1; o >>= 1) s += __shfl_xor(s, o, 32);
  const float mu = s * (1.f / 256.f);
  float sq = 0.f;
#pragma unroll
  for (int j = 0; j < 8; ++j) { const float d = v[j] - mu; sq += d * d; }
  for (int o = 16; o >= 1; o >>= 1) sq += __shfl_xor(sq, o, 32);
  const float rs = rsqrtf(sq * (1.f / 256.f) + 1e-5f);
#pragma unroll
  for (int j = 0; j < 8; ++j) {
    const int c = lane + j * 32;
    Dst[base + c] = (v[j] - mu) * rs * g[c] + be[c];
  }
}

// ---------------------------------------------------------------------------
// Multi-head attention core. QKV rows laid out [B*n, 768] = [qh|kh|vh].
// One wave per (b,h,q); scores staged in LDS; lane==channel for the
// probability-weighted V accumulation (coalesced).
// Grid must be exactly B*NHD*n/8 blocks of 256 (true for B in {2,16}, n=300).
// ---------------------------------------------------------------------------
__global__ void vd_attn_kernel(const float* __restrict__ QKV,
                               float* __restrict__ O, int B, int n) {
  const int wid = threadIdx.x >> 5, lane = threadIdx.x & 31;
  const long gid = (long)blockIdx.x * 8 + wid;
  const int q = (int)(gid % n);
  const int h = (int)((gid / n) % NHD);
  const int b = (int)(gid / ((long)n * NHD));
  const int ld = 3 * CC;

  __shared__ float sc[8][304];

  const float* Qp = QKV + (long)(b * n + q) * ld + h * HD;
  const float* Kp = QKV + (long)b * n * ld + CC + h * HD;
  const float* Vp = QKV + (long)b * n * ld + 2 * CC + h * HD;

  float qv[HD];
#pragma unroll
  for (int d = 0; d < HD; ++d) qv[d] = Qp[d];

  const float scale = 0.17677669529663687f;  // 1/sqrt(32)
  float lmax = -1e30f;
  for (int k = lane; k < n; k += 32) {
    const float* kr = Kp + (long)k * ld;
    float s = 0.f;
#pragma unroll
    for (int d = 0; d < HD; ++d) s += qv[d] * kr[d];
    s *= scale;
    sc[wid][k] = s;
    lmax = fmaxf(lmax, s);
  }
  for (int o = 16; o >= 1; o >>= 1) lmax = fmaxf(lmax, __shfl_xor(lmax, o, 32));
  float lsum = 0.f;
  for (int k = lane; k < n; k += 32) {     // each lane touches only its entries
    const float e = __expf(sc[wid][k] - lmax);
    sc[wid][k] = e;
    lsum += e;
  }
  for (int o = 16; o >= 1; o >>= 1) lsum += __shfl_xor(lsum, o, 32);
  const float inv = 1.f / lsum;

  __syncthreads();                         // make sc visible across lanes

  float acc = 0.f;
  for (int k = 0; k < n; ++k) acc += sc[wid][k] * Vp[(long)k * ld + lane];
  O[(long)(b * n + q) * CC + h * HD + lane] = acc * inv;
}

// ---------------------------------------------------------------------------
// Elementwise helpers
// ---------------------------------------------------------------------------
__global__ void vd_addqe_kernel(const float* __restrict__ X,
                                const float* __restrict__ QE,  // query_embed [300,512]
                                float* __restrict__ Dst, long nelem) {
  const long i = (long)blockIdx.x * 256 + threadIdx.x;
  if (i >= nelem) return;
  const int c = (int)(i & 255);
  const int q = (int)((i >> 8) % NQ);
  Dst[i] = X[i] + QE[q * 512 + c];
}

__global__ void vd_init_kernel(const float* __restrict__ QE,
                               float* __restrict__ outcls,
                               float* __restrict__ outloc) {
  const long i = (long)blockIdx.x * 256 + threadIdx.x;  // 4800*256 total
  const int c = (int)(i & 255);
  const long r = i >> 8;
  const int q = (int)(r % NQ);
  const float v = QE[q * 512 + CC + c];
  outloc[i] = v;
  if (r < BSZ * NQ) outcls[i] = v;
}

__global__ void vd_ref_kernel(const float* __restrict__ QE,
                              const float* __restrict__ RW,
                              const float* __restrict__ RB,
                              float* __restrict__ REF0) {
  const int t = blockIdx.x * 256 + threadIdx.x;
  if (t >= NQ * 3) return;
  const int q = t / 3, j = t % 3;
  float s = RB[j];
  for (int c = 0; c < CC; ++c) s += QE[q * 512 + c] * RW[c * 3 + j];
  REF0[t] = 1.f / (1.f + __expf(-s));
}

__global__ void vd_refout_kernel(const float* __restrict__ REF0,
                                 float* __restrict__ dst) {
  const int i = blockIdx.x * 256 + threadIdx.x;
  if (i >= B2 * NQ * 3) return;
  const int j = i % 3, q = (i / 3) % NQ;
  dst[i] = REF0[q * 3 + j];
}

// Softmax over the 16 (L*P) attention weights per (b2,q,h); in place.
__global__ void vd_awsm_kernel(float* __restrict__ AW) {
  const int t = blockIdx.x * 256 + threadIdx.x;      // 16*300*8 total, exact
  const int h = t & 7, q = (t >> 3) % NQ, b2 = t / (8 * NQ);
  float* p = AW + (long)(b2 * NQ + q) * 128 + h * 16;
  float m = -1e30f;
#pragma unroll
  for (int i = 0; i < 16; ++i) m = fmaxf(m, p[i]);
  float s = 0.f;
  float e[16];
#pragma unroll
  for (int i = 0; i < 16; ++i) { e[i] = __expf(p[i] - m); s += e[i]; }
  const float inv = 1.f / s;
#pragma unroll
  for (int i = 0; i < 16; ++i) p[i] = e[i] * inv;
}

// ---------------------------------------------------------------------------
// Deformable sampling: one wave per (b2,q,h), lane = channel.
// loc = ref0*valid_ratio + off/(W,H); bilinear gather from V with OOB zeroing.
// Writes pre-projection output in (B2,NQ,C) layout (already "transposed").
// ---------------------------------------------------------------------------
__global__ void vd_deform_kernel(const float* __restrict__ V,
                                 const float* __restrict__ OFF,
                                 const float* __restrict__ AW,
                                 const float* __restrict__ REF0,
                                 const float* __restrict__ VR,
                                 float* __restrict__ OUT) {
  const int wid = threadIdx.x >> 5, lane = threadIdx.x & 31;
  const int gid = blockIdx.x * 8 + wid;              // B2*NQ*NHD/8 blocks, exact
  const int h = gid % NHD;
  const int q = (gid / NHD) % NQ;
  const int b2 = gid / (NHD * NQ);
  const int b = b2 >> 3;

  const int Hs[4] = {64, 32, 16, 8};
  const int Ws[4] = {64, 32, 16, 8};
  const int Ss[4] = {0, 4096, 5120, 5376};

  const float* offp = OFF + (long)(b2 * NQ + q) * 512 + h * 32;  // [l][p][2]
  const float* awp  = AW  + (long)(b2 * NQ + q) * 128 + h * 16;  // [l*4+p]
  const float* vb   = V + (long)b2 * LEN * CC + h * HD + lane;

  float acc = 0.f;
#pragma unroll
  for (int l = 0; l < LV; ++l) {
    const int Hh = Hs[l], Ww = Ws[l], s0 = Ss[l];
    const float rx = REF0[q * 3 + 0] * VR[(b * LV + l) * 3 + 0];
    const float ry = REF0[q * 3 + 1] * VR[(b * LV + l) * 3 + 1];
#pragma unroll
    for (int p = 0; p < PP; ++p) {
      const float ox = offp[(l * 4 + p) * 2 + 0];
      const float oy = offp[(l * 4 + p) * 2 + 1];
      const float lx = rx + ox / (float)Ww;
      const float ly = ry + oy / (float)Hh;
      const float x = lx * Ww - 0.5f, y = ly * Hh - 0.5f;
      const float x0f = floorf(x), y0f = floorf(y);
      const float dx = x - x0f, dy = y - y0f;
      const int x0 = (int)x0f, y0 = (int)y0f;
      float sample = 0.f;
      {
        const int xi = x0, yi = y0;
        if (xi >= 0 && xi < Ww && yi >= 0 && yi < Hh)
          sample += (1.f - dx) * (1.f - dy) * vb[(long)(s0 + yi * Ww + xi) * CC];
      }
      {
        const int xi = x0 + 1, yi = y0;
        if (xi >= 0 && xi < Ww && yi >= 0 && yi < Hh)
          sample += dx * (1.f - dy) * vb[(long)(s0 + yi * Ww + xi) * CC];
      }
      {
        const int xi = x0, yi = y0 + 1;
        if (xi >= 0 && xi < Ww && yi >= 0 && yi < Hh)
          sample += (1.f - dx) * dy * vb[(long)(s0 + yi * Ww + xi) * CC];
      }
      {
        const int xi = x0 + 1, yi = y0 + 1;
        if (xi >= 0 && xi < Ww && yi >= 0 && yi < Hh)
          sample += dx * dy * vb[(long)(s0 + yi * Ww + xi) * CC];
      }
      acc += awp[l * 4 + p] * sample;
    }
  }
  OUT[(long)(b2 * NQ + q) * CC + h * HD + lane] = acc;
}

// ---------------------------------------------------------------------------
// tw pipeline: per-row dot with tw.w; softmax over t; weighted sum -> tgt2_cls
// ---------------------------------------------------------------------------
__global__ void vd_twdot_kernel(const float* __restrict__ X,
                                const float* __restrict__ w,
                                const float* __restrict__ b,
                                float* __restrict__ TWL) {
  const int wid = threadIdx.x >> 5, lane = threadIdx.x & 31;
  const int r = blockIdx.x * 8 + wid;                // 4800 rows exact
  float s = 0.f;
#pragma unroll
  for (int j = 0; j < 8; ++j) { const int c = lane + j * 32; s += X[(long)r * CC + c] * w[c]; }
  for (int o = 16; o >= 1; o >>= 1) s += __shfl_xor(s, o, 32);
  if (lane == 0) TWL[r] = s + b[0];
}

__global__ void vd_twsm_kernel(const float* __restrict__ TWL,
                               float* __restrict__ TWS) {
  const int t = blockIdx.x * 256 + threadIdx.x;
  if (t >= BSZ * NQ) return;
  const int b = t / NQ, q = t % NQ;
  float v[TT], m = -1e30f;
#pragma unroll
  for (int tt = 0; tt < TT; ++tt) { v[tt] = TWL[(b * TT + tt) * NQ + q]; m = fmaxf(m, v[tt]); }
  float s = 0.f;
#pragma unroll
  for (int tt = 0; tt < TT; ++tt) { v[tt] = __expf(v[tt] - m); s += v[tt]; }
  const float inv = 1.f / s;
#pragma unroll
  for (int tt = 0; tt < TT; ++tt) TWS[(b * TT + tt) * NQ + q] = v[tt] * inv;
}

__global__ void vd_cls2_kernel(const float* __restrict__ T2LOC,
                               const float* __restrict__ TWS,
                               float* __restrict__ CLS2) {
  const int i = blockIdx.x * 256 + threadIdx.x;      // 2*300*256 exact
  const int c = i & 255;
  const int r = i >> 8;
  const int q = r % NQ, b = r / NQ;
  float s = 0.f;
#pragma unroll
  for (int t = 0; t < TT; ++t) {
    const int b2 = b * TT + t;
    s += T2LOC[((long)(b2 * NQ + q) << 8) + c] * TWS[b2 * NQ + q];
  }
  CLS2[i] = s;
}

// ---------------------------------------------------------------------------
// Host side
// ---------------------------------------------------------------------------

// Per-layer param-leaf indices (jax pytree: dict keys sorted; lin={b,w}, ln={b,g})
enum {
  CA_AW_B = 0, CA_AW_W, CA_O_B, CA_O_W, CA_OFF_B, CA_OFF_W, CA_V_B, CA_V_W,
  FF1_CLS_B, FF1_CLS_W, FF1_LOC_B, FF1_LOC_W, FF2_CLS_B, FF2_CLS_W, FF2_LOC_B, FF2_LOC_W,
  LN1_CLS_B, LN1_CLS_G, LN1_LOC_B, LN1_LOC_G, LN2_CLS_B, LN2_CLS_G, LN2_LOC_B, LN2_LOC_G,
  LN3_CLS_B, LN3_CLS_G, LN3_LOC_B, LN3_LOC_G,
  SA_CLS_IN_B, SA_CLS_IN_W, SA_CLS_OUT_B, SA_CLS_OUT_W,
  SA_LOC_IN_B, SA_LOC_IN_W, SA_LOC_OUT_B, SA_LOC_OUT_W,
  TW_B, TW_W, NUM_LEAVES  // 38
};

static inline void launch_gemm(hipStream_t s, const float* A, int lda,
                               const float* W, int ldw, const float* bias,
                               float* D, int ldd, int M, int K, int N,
                               int relu, const unsigned char* rowmask) {
  dim3 grid((M + 31) / 32, (N + 63) / 64);
  vd_gemm_kernel<<<grid, dim3(256), 0, s>>>(A, lda, W, ldw, bias, D, ldd,
                                            M, K, N, relu, rowmask);
}

extern "C" void kernel_launch(void* const* d_in, const int* in_sizes, int n_in,
                              void* d_out, int out_size, void* d_ws, size_t ws_size,
                              hipStream_t stream) {
  (void)in_sizes; (void)n_in; (void)out_size; (void)ws_size;

  const float* memory  = (const float*)d_in[0];
  const float* vratios = (const float*)d_in[3];
  const unsigned char* mask = (const unsigned char*)d_in[4];
  const float* qe      = (const float*)d_in[5];  // query_embed [300,512]
  const int PBASE = 6;
  const float* ref_b = (const float*)d_in[PBASE + NL * NUM_LEAVES + 0];
  const float* ref_w = (const float*)d_in[PBASE + NL * NUM_LEAVES + 1];

  float* ws  = (float*)d_ws;
  float* out = (float*)d_out;

  // Workspace layout (floats)
  const long oV      = 0;                         // 16*5440*256
  const long oQKV    = oV      + (long)B2 * LEN * CC;   // also FFN hidden (ld 1024)
  const long oQIN    = oQKV    + (long)B2 * NQ * FFD;
  const long oT2PRE  = oQIN    + (long)B2 * NQ * CC;
  const long oT2LOC  = oT2PRE  + (long)B2 * NQ * CC;
  const long oTMP1   = oT2LOC  + (long)B2 * NQ * CC;
  const long oTMP2   = oTMP1   + (long)B2 * NQ * CC;
  const long oOFF    = oTMP2   + (long)B2 * NQ * CC;    // 16*300*512
  const long oAW     = oOFF    + (long)B2 * NQ * 512;   // 16*300*128
  const long oOUTCLS = oAW     + (long)B2 * NQ * 128;
  const long oOUTLOC = oOUTCLS + (long)BSZ * NQ * CC;
  const long oCLS2   = oOUTLOC + (long)B2 * NQ * CC;
  const long oREF0   = oCLS2   + (long)BSZ * NQ * CC;
  const long oTWL    = oREF0   + NQ * 3;
  const long oTWS    = oTWL    + B2 * NQ;

  float* V      = ws + oV;
  float* QKV    = ws + oQKV;
  float* QIN    = ws + oQIN;
  float* T2PRE  = ws + oT2PRE;
  float* T2LOC  = ws + oT2LOC;
  float* TMP1   = ws + oTMP1;
  float* TMP2   = ws + oTMP2;
  float* OFFB   = ws + oOFF;
  float* AWB    = ws + oAW;
  float* OUTCLS = ws + oOUTCLS;
  float* OUTLOC = ws + oOUTLOC;
  float* CLS2   = ws + oCLS2;
  float* REF0   = ws + oREF0;
  float* TWL    = ws + oTWL;
  float* TWS    = ws + oTWS;

  // Output layout (floats): ic[6,2,300,256] | il[6,2,8,300,256] | ref[2,8,300,3] | ir[6,...]
  const long IC_OFF  = 0;
  const long IC_SZ   = (long)BSZ * NQ * CC;            // 153600
  const long IL_OFF  = IC_OFF + NL * IC_SZ;
  const long IL_SZ   = (long)B2 * NQ * CC;             // 1228800
  const long REF_OFF = IL_OFF + NL * IL_SZ;
  const long REF_SZ  = (long)B2 * NQ * 3;              // 14400
  const long IR_OFF  = REF_OFF + REF_SZ;

  const int RLOC = B2 * NQ;    // 4800
  const int RCLS = BSZ * NQ;   // 600

  // --- init: tgt broadcast, ref projection + sigmoid, ref/ir outputs ---
  vd_init_kernel<<<(RLOC * CC) / 256, 256, 0, stream>>>(qe, OUTCLS, OUTLOC);
  vd_ref_kernel<<<(NQ * 3 + 255) / 256, 256, 0, stream>>>(qe, ref_w, ref_b, REF0);
  vd_refout_kernel<<<(B2 * NQ * 3 + 255) / 256, 256, 0, stream>>>(REF0, out + REF_OFF);
  for (int l = 0; l < NL; ++l)
    hipMemcpyAsync(out + IR_OFF + (long)l * REF_SZ, out + REF_OFF,
                   REF_SZ * sizeof(float), hipMemcpyDeviceToDevice, stream);

  for (int L = 0; L < NL; ++L) {
    const int base = PBASE + L * NUM_LEAVES;
    #define LP(leaf) ((const float*)d_in[base + (leaf)])

    // ---- cls self-attention (B=2, 600 rows) ----
    vd_addqe_kernel<<<(RCLS * CC) / 256, 256, 0, stream>>>(OUTCLS, qe, QIN, (long)RCLS * CC);
    launch_gemm(stream, QIN, CC, LP(SA_CLS_IN_W), 3 * CC, LP(SA_CLS_IN_B),
                QKV, 3 * CC, RCLS, CC, 2 * CC, 0, nullptr);            // qh|kh
    launch_gemm(stream, OUTCLS, CC, LP(SA_CLS_IN_W) + 2 * CC, 3 * CC,
                LP(SA_CLS_IN_B) + 2 * CC, QKV + 2 * CC, 3 * CC,
                RCLS, CC, CC, 0, nullptr);                              // vh
    vd_attn_kernel<<<(BSZ * NHD * NQ) / 8, 256, 0, stream>>>(QKV, TMP1, BSZ, NQ);
    launch_gemm(stream, TMP1, CC, LP(SA_CLS_OUT_W), CC, LP(SA_CLS_OUT_B),
                TMP2, CC, RCLS, CC, CC, 0, nullptr);
    vd_ln_kernel<<<(RCLS + 7) / 8, 256, 0, stream>>>(OUTCLS, TMP2,
                LP(LN2_CLS_G), LP(LN2_CLS_B), OUTCLS, RCLS);

    // ---- loc self-attention (B=16, 4800 rows; layer-0 3D path is
    //      equivalent since all t-slices are identical copies) ----
    vd_addqe_kernel<<<(RLOC * CC) / 256, 256, 0, stream>>>(OUTLOC, qe, QIN, (long)RLOC * CC);
    launch_gemm(stream, QIN, CC, LP(SA_LOC_IN_W), 3 * CC, LP(SA_LOC_IN_B),
                QKV, 3 * CC, RLOC, CC, 2 * CC, 0, nullptr);
    launch_gemm(stream, OUTLOC, CC, LP(SA_LOC_IN_W) + 2 * CC, 3 * CC,
                LP(SA_LOC_IN_B) + 2 * CC, QKV + 2 * CC, 3 * CC,
                RLOC, CC, CC, 0, nullptr);
    vd_attn_kernel<<<(B2 * NHD * NQ) / 8, 256, 0, stream>>>(QKV, TMP1, B2, NQ);
    launch_gemm(stream, TMP1, CC, LP(SA_LOC_OUT_W), CC, LP(SA_LOC_OUT_B),
                TMP2, CC, RLOC, CC, CC, 0, nullptr);
    vd_ln_kernel<<<(RLOC + 7) / 8, 256, 0, stream>>>(OUTLOC, TMP2,
                LP(LN2_LOC_G), LP(LN2_LOC_B), OUTLOC, RLOC);

    // ---- deformable cross-attention ----
    vd_addqe_kernel<<<(RLOC * CC) / 256, 256, 0, stream>>>(OUTLOC, qe, QIN, (long)RLOC * CC);
    launch_gemm(stream, memory, CC, LP(CA_V_W), CC, LP(CA_V_B),
                V, CC, B2 * LEN, CC, CC, 0, mask);                      // value proj + mask
    launch_gemm(stream, QIN, CC, LP(CA_OFF_W), 512, LP(CA_OFF_B),
                OFFB, 512, RLOC, CC, 512, 0, nullptr);                  // offsets
    launch_gemm(stream, QIN, CC, LP(CA_AW_W), 128, LP(CA_AW_B),
                AWB, 128, RLOC, CC, 128, 0, nullptr);                   // aw logits
    vd_awsm_kernel<<<(B2 * NQ * NHD) / 256, 256, 0, stream>>>(AWB);
    vd_deform_kernel<<<(B2 * NQ * NHD) / 8, 256, 0, stream>>>(V, OFFB, AWB,
                                                              REF0, vratios, T2PRE);
    launch_gemm(stream, T2PRE, CC, LP(CA_O_W), CC, LP(CA_O_B),
                T2LOC, CC, RLOC, CC, CC, 0, nullptr);                   // output proj
    vd_ln_kernel<<<(RLOC + 7) / 8, 256, 0, stream>>>(OUTLOC, T2LOC,
                LP(LN1_LOC_G), LP(LN1_LOC_B), OUTLOC, RLOC);

    // ---- FFN loc ----
    launch_gemm(stream, OUTLOC, CC, LP(FF1_LOC_W), FFD, LP(FF1_LOC_B),
                QKV, FFD, RLOC, CC, FFD, 1, nullptr);
    launch_gemm(stream, QKV, FFD, LP(FF2_LOC_W), CC, LP(FF2_LOC_B),
                TMP1, CC, RLOC, FFD, CC, 0, nullptr);
    vd_ln_kernel<<<(RLOC + 7) / 8, 256, 0, stream>>>(OUTLOC, TMP1,
                LP(LN3_LOC_G), LP(LN3_LOC_B), OUTLOC, RLOC);

    // ---- temporal weighting -> cls update ----
    vd_twdot_kernel<<<RLOC / 8, 256, 0, stream>>>(OUTLOC, LP(TW_W), LP(TW_B), TWL);
    vd_twsm_kernel<<<(RCLS + 255) / 256, 256, 0, stream>>>(TWL, TWS);
    vd_cls2_kernel<<<(RCLS * CC) / 256, 256, 0, stream>>>(T2LOC, TWS, CLS2);
    vd_ln_kernel<<<(RCLS + 7) / 8, 256, 0, stream>>>(OUTCLS, CLS2,
                LP(LN1_CLS_G), LP(LN1_CLS_B), OUTCLS, RCLS);

    // ---- FFN cls ----
    launch_gemm(stream, OUTCLS, CC, LP(FF1_CLS_W), FFD, LP(FF1_CLS_B),
                QKV, FFD, RCLS, CC, FFD, 1, nullptr);
    launch_gemm(stream, QKV, FFD, LP(FF2_CLS_W), CC, LP(FF2_CLS_B),
                TMP1, CC, RCLS, FFD, CC, 0, nullptr);
    vd_ln_kernel<<<(RCLS + 7) / 8, 256, 0, stream>>>(OUTCLS, TMP1,
                LP(LN3_CLS_G), LP(LN3_CLS_B), OUTCLS, RCLS);

    // ---- emit layer outputs ----
    hipMemcpyAsync(out + IC_OFF + (long)L * IC_SZ, OUTCLS,
                   IC_SZ * sizeof(float), hipMemcpyDeviceToDevice, stream);
    hipMemcpyAsync(out + IL_OFF + (long)L * IL_SZ, OUTLOC,
                   IL_SZ * sizeof(float), hipMemcpyDeviceToDevice, stream);
    #undef LP
  }
}